// LSTMModel_76888504533690
// MI455X (gfx1250) — compile-verified
//
#include <hip/hip_runtime.h>
#include <math.h>

typedef __attribute__((ext_vector_type(2))) float v2f;
typedef __attribute__((ext_vector_type(8))) float v8f;

#define SEQ   4096
#define BATCH 1024
#define INSZ  9
#define XLD   48   // padded leading dim of xproj (3 N-tiles of 16)
#define H5LD  8    // padded leading dim of final-layer h sequence

struct LstmParams {
  const float* Wih[6];
  const float* Whh[6];
  const float* bih[6];
  const float* bhh[6];
};

// ---------------------------------------------------------------------------
// Kernel 1: xproj = x[:,1023,:] @ Wih0^T + (bih0 + bhh0)   -> (4096 x 48)
// f32 WMMA 16x16x4. One wave per (16-row seq tile, 16-col gate tile).
// Loads are branch-free: clamp the index, select the value.
// ---------------------------------------------------------------------------
__global__ __launch_bounds__(32) void xproj0_wmma_kernel(
    const float* __restrict__ x,     // (SEQ, BATCH, 9)
    const float* __restrict__ Wih,   // (36, 9) row-major
    const float* __restrict__ bih,   // (36)
    const float* __restrict__ bhh,   // (36)
    float* __restrict__ xproj)       // (SEQ, XLD)
{
  const int mtile = blockIdx.x;        // 0..255 (seq tiles)
  const int ntile = blockIdx.y;        // 0..2   (gate tiles)
  const int lane  = threadIdx.x;       // 0..31
  const int nloc  = lane & 15;         // N (and A-side M) index within tile
  const int khalf = lane >> 4;         // selects K pair {0,1} vs {2,3}
  const int gn    = ntile * 16 + nloc; // gate column 0..47
  const int gns   = (gn < 36) ? gn : 0;

  // C initialized with bias broadcast along M (bias is per-gate-column)
  const float braw = bih[gns] + bhh[gns];
  const float bias = (gn < 36) ? braw : 0.0f;
  v8f c;
#pragma unroll
  for (int r = 0; r < 8; ++r) c[r] = bias;

  const int m = mtile * 16 + nloc;     // A-matrix row (seq index) for this lane
  const float* xrow = x + (size_t)m * (BATCH * INSZ) + (size_t)(BATCH - 1) * INSZ;

#pragma unroll
  for (int k0 = 0; k0 < 12; k0 += 4) {           // K padded 9 -> 12
    const int ka  = k0 + khalf * 2;
    const int k0i = (ka + 0 < INSZ) ? ka + 0 : 0;
    const int k1i = (ka + 1 < INSZ) ? ka + 1 : 0;
    const float ax0 = xrow[k0i], ax1 = xrow[k1i];
    const float bw0 = Wih[gns * INSZ + k0i], bw1 = Wih[gns * INSZ + k1i];
    v2f a, b;
    // A: 16x4 f32, lane%16 = M, VGPR0/1 hold K = khalf*2 + {0,1}
    a[0] = (ka + 0 < INSZ) ? ax0 : 0.0f;
    a[1] = (ka + 1 < INSZ) ? ax1 : 0.0f;
    // B: 4x16 f32, B[k][n] = Wih[n][k]
    b[0] = (gn < 36 && ka + 0 < INSZ) ? bw0 : 0.0f;
    b[1] = (gn < 36 && ka + 1 < INSZ) ? bw1 : 0.0f;
    c = __builtin_amdgcn_wmma_f32_16x16x4_f32(false, a, false, b,
                                              (short)0, c, false, false);
  }

  // D layout: VGPR r holds M = r + (lane>=16 ? 8 : 0), N = lane%16
  const int mbase = mtile * 16 + khalf * 8;
#pragma unroll
  for (int r = 0; r < 8; ++r)
    xproj[(size_t)(mbase + r) * XLD + gn] = c[r];
}

// sigmoid(x) = 1 - 1*rcp(1 + e^{1x});  tanh(x) = 1 - 2*rcp(1 + e^{2x})
__device__ __forceinline__ float act_unified(float x, float s) {
  return 1.0f - s * __builtin_amdgcn_rcpf(1.0f + __expf(s * x));
}
__device__ __forceinline__ float fast_tanh(float x) {
  return 1.0f - 2.0f * __builtin_amdgcn_rcpf(1.0f + __expf(2.0f * x));
}

// ---------------------------------------------------------------------------
// Kernel 2: 6 LSTM layers software-pipelined across one workgroup.
// Wave-group l handles layer l at time t = s - l; h exchanged via LDS
// double buffers; 2 workgroup barriers per global step. Layer-0's xproj
// stream is prefetched 4 steps ahead in registers to keep L2 latency off
// the 4096-step dependence chain.
// ---------------------------------------------------------------------------
__global__ __launch_bounds__(256) void lstm_pipeline_kernel(
    LstmParams P,
    const float* __restrict__ xproj,  // (SEQ, XLD), layer-0 input proj + bias
    float* __restrict__ h5)           // (SEQ, H5LD), layer-5 h sequence
{
  __shared__ float h_sh[6][2][16];    // [layer][slot][j], cols H..15 stay 0
  __shared__ float gate_sh[6][36];

  const int tid = threadIdx.x;
  int l, g;
  if (tid < 64)       { l = 0; g = tid; }
  else if (tid < 128) { l = 1; g = tid - 64; }
  else                { l = 2 + ((tid - 128) >> 5); g = (tid - 128) & 31; }

  const int din = (l <= 2) ? 9 : 6;   // layer input widths {9,9,9,6,6,6}
  const int Hh  = (l <= 1) ? 9 : 6;   // hidden widths      {9,9,6,6,6,6}
  const int G   = 4 * Hh;
  const bool inrole = (g < G);
  const int gclamp = (g < 36) ? g : 35;

  // Cache weight rows in registers (fixed-9 arrays, zero-padded tails).
  float win[9], wrec[9];
#pragma unroll
  for (int k = 0; k < 9; ++k)
    win[k] = (inrole && l > 0 && k < din) ? P.Wih[l][g * din + k] : 0.0f;
#pragma unroll
  for (int j = 0; j < 9; ++j)
    wrec[j] = (inrole && j < Hh) ? P.Whh[l][g * Hh + j] : 0.0f;
  const float bias = (inrole && l > 0) ? (P.bih[l][g] + P.bhh[l][g]) : 0.0f;

  // gate section: 0=i, 1=f, 2=g(tanh), 3=o ; unified activation constant
  const int sect = !inrole ? 0 : (g >= 3 * Hh) ? 3 : (g >= 2 * Hh) ? 2
                                                   : (g >= Hh) ? 1 : 0;
  const float sgk = (sect == 2) ? 2.0f : 1.0f;
  float creg = 0.0f;                  // cell state for thread g (< Hh)

  // 4-deep xproj prefetch ring (layer-0 waves only; l is wave-uniform)
  float xp0 = 0.0f, xp1 = 0.0f, xp2 = 0.0f, xp3 = 0.0f;
  if (l == 0) {
    xp0 = xproj[0 * XLD + gclamp];
    xp1 = xproj[1 * XLD + gclamp];
    xp2 = xproj[2 * XLD + gclamp];
    xp3 = xproj[3 * XLD + gclamp];
  }

  if (tid < 192) ((float*)h_sh)[tid] = 0.0f;   // h(-1) = 0, padding = 0
  __syncthreads();

#pragma unroll 4
  for (int s = 0; s < SEQ + 8; ++s) {          // 4104 steps (mult. of 4)
    const int tcur = s - l;
    const bool active = inrole && (tcur >= 0) && (tcur < SEQ);
    const int rd = (s + 1) & 1;       // slot written at step s-1
    const int wr = s & 1;

    // Issue prefetch for t = s+4 early (completes ~4 steps later).
    float xp_new = 0.0f;
    if (l == 0) {
      const int tf = (s + 4 < SEQ) ? (s + 4) : (SEQ - 1);
      xp_new = xproj[tf * XLD + gclamp];
    }

    // Phase 1: gate pre-activations + nonlinearity
    if (active) {
      float acc;
      if (l == 0) {
        acc = xp0;                    // xproj[tcur], bias already folded in
      } else {
        acc = bias;
        const float* hin = h_sh[l - 1][rd];   // h_{l-1}(tcur)
#pragma unroll
        for (int k = 0; k < 9; ++k) acc += win[k] * hin[k];
      }
      const float* hown = h_sh[l][rd];        // h_l(tcur - 1)
#pragma unroll
      for (int j = 0; j < 9; ++j) acc += wrec[j] * hown[j];
      gate_sh[l][g] = act_unified(acc, sgk);
    }
    __syncthreads();

    // Phase 2: cell/hidden update by threads g < Hh
    if (active && g < Hh) {
      const float iv = gate_sh[l][g];
      const float fv = gate_sh[l][Hh + g];
      const float gv = gate_sh[l][2 * Hh + g];
      const float ov = gate_sh[l][3 * Hh + g];
      creg = fv * creg + iv * gv;
      const float hv = ov * fast_tanh(creg);
      h_sh[l][wr][g] = hv;
      if (l == 5) h5[tcur * H5LD + g] = hv;
    }
    __syncthreads();

    xp0 = xp1; xp1 = xp2; xp2 = xp3; xp3 = xp_new;   // renamed by unroll
  }
}

// ---------------------------------------------------------------------------
// Kernel 3: z[t][n] = max(-(h5[t] . fc_w[n] + fc_b[n]) / 0.1, -1)
// ---------------------------------------------------------------------------
__global__ __launch_bounds__(256) void fc_z_kernel(
    const float* __restrict__ h5,     // (SEQ, H5LD)
    const float* __restrict__ fc_w,   // (6, 6) row-major
    const float* __restrict__ fc_b,   // (6)
    float* __restrict__ out)          // (SEQ, 6)
{
  const int idx = blockIdx.x * 256 + threadIdx.x;
  if (idx >= SEQ * 6) return;
  const int t = idx / 6, n = idx % 6;
  float acc = fc_b[n];
#pragma unroll
  for (int j = 0; j < 6; ++j) acc += h5[t * H5LD + j] * fc_w[n * 6 + j];
  out[idx] = fmaxf(-10.0f * acc, -1.0f);
}

// ---------------------------------------------------------------------------
extern "C" void kernel_launch(void* const* d_in, const int* in_sizes, int n_in,
                              void* d_out, int out_size, void* d_ws, size_t ws_size,
                              hipStream_t stream) {
  (void)in_sizes; (void)n_in; (void)out_size; (void)ws_size;

  const float* x = (const float*)d_in[0];
  LstmParams P;
  for (int l = 0; l < 6; ++l) {
    P.Wih[l] = (const float*)d_in[1 + 4 * l];
    P.Whh[l] = (const float*)d_in[2 + 4 * l];
    P.bih[l] = (const float*)d_in[3 + 4 * l];
    P.bhh[l] = (const float*)d_in[4 + 4 * l];
  }
  const float* fc_w = (const float*)d_in[25];
  const float* fc_b = (const float*)d_in[26];

  float* xproj = (float*)d_ws;                   // 4096*48 f32 = 768 KB
  float* h5    = xproj + (size_t)SEQ * XLD;      // 4096*8  f32 = 128 KB
  float* out   = (float*)d_out;

  xproj0_wmma_kernel<<<dim3(256, 3), 32, 0, stream>>>(
      x, P.Wih[0], P.bih[0], P.bhh[0], xproj);
  lstm_pipeline_kernel<<<dim3(1), 256, 0, stream>>>(P, xproj, h5);
  fc_z_kernel<<<dim3(96), 256, 0, stream>>>(h5, fc_w, fc_b, out);
}